// HJRLConv_67619965108616
// MI455X (gfx1250) — compile-verified
//
#include <hip/hip_runtime.h>

// CDNA5 (gfx1250) GCN conv layer:
//   support = X @ W            (f32 WMMA 16x16x4, W staged in LDS)
//   out[dst] += val * support[src]   (1 wave per edge, f32 L2 atomics)
//   out = relu(out)
//
// d_ws layout: support[N_NODES * 128] floats (51.2 MB).

typedef __attribute__((ext_vector_type(2))) float v2f;
typedef __attribute__((ext_vector_type(8))) float v8f;

#define DFEAT 128

// ---------------------------------------------------------------------------
// Kernel 1: support = X @ W   via V_WMMA_F32_16X16X4_F32
// One wave (32 lanes) computes a 16x128 strip: 8 tiles of 16x16, K=128 in
// steps of 4. 8 waves / block; W (64KB) cooperatively staged into LDS.
// A-frag layout (ISA 32-bit 16x4 table): lanes 0-15 hold K=k,k+1 for M=lane;
// lanes 16-31 hold K=k+2,k+3 for M=lane-16. B mirrors (row striped in lanes).
// C/D layout: VGPR v, lanes 0-15 => M=v, N=lane; lanes 16-31 => M=v+8.
// ---------------------------------------------------------------------------
__global__ void __launch_bounds__(256)
gcn_gemm_wmma_f32(const float* __restrict__ X,
                  const float* __restrict__ W,
                  float* __restrict__ support,
                  int n_rows) {
    __shared__ float Wlds[DFEAT * DFEAT];   // 64 KB

    // Cooperative load of W into LDS (coalesced b32 -> b128 merges).
    for (int i = threadIdx.x; i < DFEAT * DFEAT; i += 256)
        Wlds[i] = W[i];
    __syncthreads();

    const int lane   = threadIdx.x & 31;
    const int waveIb = threadIdx.x >> 5;
    const int waveId = blockIdx.x * 8 + waveIb;
    const int m0     = waveId * 16;
    if (m0 >= n_rows) return;        // full waves only past here: EXEC == ~0

    const int half = lane >> 4;      // 0: K=k,k+1  1: K=k+2,k+3
    const int l16  = lane & 15;

    v8f acc[8];
#pragma unroll
    for (int t = 0; t < 8; ++t)
        acc[t] = (v8f){0.f, 0.f, 0.f, 0.f, 0.f, 0.f, 0.f, 0.f};

    const float* xrow = X + (size_t)(m0 + l16) * DFEAT;

    for (int kk = 0; kk < DFEAT; kk += 4) {
        const int ka = kk + half * 2;
        v2f a;
        a.x = xrow[ka];              // contiguous pair -> global_load_b64
        a.y = xrow[ka + 1];
#pragma unroll
        for (int t = 0; t < 8; ++t) {
            const int col = t * 16 + l16;
            v2f b;
            b.x = Wlds[(size_t)ka * DFEAT + col];
            b.y = Wlds[(size_t)(ka + 1) * DFEAT + col];
            acc[t] = __builtin_amdgcn_wmma_f32_16x16x4_f32(
                /*neg_a=*/false, a, /*neg_b=*/false, b,
                /*c_mod=*/(short)0, acc[t],
                /*reuse_a=*/false, /*reuse_b=*/false);
        }
    }

    // Store 16x128 strip. Half-wave writes 64B contiguous per (tile,row).
    const int rbase = m0 + half * 8;
#pragma unroll
    for (int t = 0; t < 8; ++t) {
#pragma unroll
        for (int v = 0; v < 8; ++v) {
            support[(size_t)(rbase + v) * DFEAT + t * 16 + l16] = acc[t][v];
        }
    }
}

// ---------------------------------------------------------------------------
// Kernel 2: edge scatter. One wave per edge: 32 lanes x float4 = 512B row.
// support + out both fit in the 192MB L2, so gathers hit L2 and the f32
// atomic adds resolve at L2 atomic units.
// ---------------------------------------------------------------------------
__global__ void __launch_bounds__(256)
gcn_edge_scatter(const float* __restrict__ support,
                 const float* __restrict__ edge_vals,
                 const int*   __restrict__ edge_src,
                 const int*   __restrict__ edge_dst,
                 float* __restrict__ out,
                 int n_edges) {
    const int wid  = (int)((blockIdx.x * 256u + threadIdx.x) >> 5);
    const int lane = threadIdx.x & 31;
    if (wid >= n_edges) return;

    const int   s = edge_src[wid];   // wave-uniform -> scalarized
    const int   d = edge_dst[wid];
    const float v = edge_vals[wid];

    const float4 m =
        *(reinterpret_cast<const float4*>(support + (size_t)s * DFEAT) + lane);

    float* o = out + (size_t)d * DFEAT + lane * 4;
    atomicAdd(o + 0, m.x * v);       // global_atomic_add_f32
    atomicAdd(o + 1, m.y * v);
    atomicAdd(o + 2, m.z * v);
    atomicAdd(o + 3, m.w * v);
}

// ---------------------------------------------------------------------------
// Kernel 3: ReLU (neg_slope == 0), float4 vectorized.
// ---------------------------------------------------------------------------
__global__ void __launch_bounds__(256)
gcn_relu4(float* __restrict__ out, int n4) {
    const int i = blockIdx.x * 256 + threadIdx.x;
    if (i >= n4) return;
    float4* p = reinterpret_cast<float4*>(out) + i;
    float4 x = *p;
    x.x = fmaxf(x.x, 0.f);
    x.y = fmaxf(x.y, 0.f);
    x.z = fmaxf(x.z, 0.f);
    x.w = fmaxf(x.w, 0.f);
    *p = x;
}

// ---------------------------------------------------------------------------
extern "C" void kernel_launch(void* const* d_in, const int* in_sizes, int n_in,
                              void* d_out, int out_size, void* d_ws, size_t ws_size,
                              hipStream_t stream) {
    const float* X    = (const float*)d_in[0];   // [N, 128]
    const float* W    = (const float*)d_in[1];   // [128, 128]
    const float* vals = (const float*)d_in[2];   // [E]
    const int*   src  = (const int*)d_in[3];     // [E]
    const int*   dst  = (const int*)d_in[4];     // [E]
    float*       out  = (float*)d_out;           // [N, 128]
    float*       sup  = (float*)d_ws;            // [N, 128] scratch (51.2 MB)

    const int n_nodes = in_sizes[0] / DFEAT;
    const int n_edges = in_sizes[2];

    // 1) support = X @ W  (N_NODES divisible by 16 -> exact tiling)
    const int gemm_waves  = (n_nodes + 15) / 16;
    const int gemm_blocks = (gemm_waves + 7) / 8;
    gcn_gemm_wmma_f32<<<gemm_blocks, 256, 0, stream>>>(X, W, sup, n_nodes);

    // 2) out = 0  (graph-capturable async memset)
    hipMemsetAsync(d_out, 0, (size_t)out_size * sizeof(float), stream);

    // 3) out[dst] += val * support[src]   (8 edges/block)
    const int sc_blocks = (n_edges + 7) / 8;
    gcn_edge_scatter<<<sc_blocks, 256, 0, stream>>>(sup, vals, src, dst, out, n_edges);

    // 4) relu
    const int n4 = out_size / 4;
    gcn_relu4<<<(n4 + 255) / 256, 256, 0, stream>>>(out, n4);
}